// MinibatchDiscriminationV1_18451179504181
// MI455X (gfx1250) — compile-verified
//
#include <hip/hip_runtime.h>

// fp32 WMMA fragment types (wave32): A/B 16x4 & 4x16 f32 = 2 VGPRs, C/D 16x16 f32 = 8 VGPRs
typedef __attribute__((ext_vector_type(2))) float v2f;
typedef __attribute__((ext_vector_type(8))) float v8f;

constexpr int kN    = 256;   // batch
constexpr int kA    = 1024;  // input features
constexpr int kB    = 128;   // minibatch-feature count
constexpr int kC    = 16;    // kernel dim
constexpr int kBC   = kB * kC;   // 2048
constexpr int kOutW = kA + kB;   // 1152

// ---------------------------------------------------------------------------
// Kernel 1: act[256,2048] = inp[256,1024] @ theta[1024,2048], fp32 WMMA.
// Register-blocked: each wave computes a 16x64 tile (4 accumulators), reusing
// one A fragment across 4 B tiles per k-step. Block = 8 waves = 512 columns.
// grid = (2048/512, 256/16) = (4, 16).
// ---------------------------------------------------------------------------
__global__ __launch_bounds__(256) void gemm_f32_wmma(const float* __restrict__ inp,
                                                     const float* __restrict__ theta,
                                                     float* __restrict__ act) {
    const int wave = threadIdx.x >> 5;   // 0..7
    const int lane = threadIdx.x & 31;
    const int hs   = lane >> 4;          // half-select: 0 for lanes 0-15, 1 for 16-31
    const int lr   = lane & 15;

    const int m0 = blockIdx.y * 16;
    const int n0 = blockIdx.x * 512 + wave * 64;

    const float* Ap = inp + (size_t)(m0 + lr) * kA;  // row of A owned by this lane
    const float* Bp = theta + n0 + lr;               // base column of B for this lane

    v8f acc0 = {0.f, 0.f, 0.f, 0.f, 0.f, 0.f, 0.f, 0.f};
    v8f acc1 = acc0, acc2 = acc0, acc3 = acc0;

#pragma unroll 2
    for (int k = 0; k < kA; k += 4) {
        const int ka = k + 2 * hs;
        // A 16x4: lanes 0-15 carry K={k,k+1}, lanes 16-31 carry K={k+2,k+3}
        v2f a;
        a.x = Ap[ka];
        a.y = Ap[ka + 1];
        // B 4x16 per tile: VGPR0 rows K={k,k+2}, VGPR1 rows K={k+1,k+3} by half
        const float* b_lo = Bp + (size_t)ka * kBC;
        const float* b_hi = Bp + (size_t)(ka + 1) * kBC;
        v2f b0, b1, b2, b3;
        b0.x = b_lo[0];  b0.y = b_hi[0];
        b1.x = b_lo[16]; b1.y = b_hi[16];
        b2.x = b_lo[32]; b2.y = b_hi[32];
        b3.x = b_lo[48]; b3.y = b_hi[48];

        acc0 = __builtin_amdgcn_wmma_f32_16x16x4_f32(false, a, false, b0, (short)0, acc0, false, false);
        acc1 = __builtin_amdgcn_wmma_f32_16x16x4_f32(false, a, false, b1, (short)0, acc1, false, false);
        acc2 = __builtin_amdgcn_wmma_f32_16x16x4_f32(false, a, false, b2, (short)0, acc2, false, false);
        acc3 = __builtin_amdgcn_wmma_f32_16x16x4_f32(false, a, false, b3, (short)0, acc3, false, false);
    }

    // D 16x16 f32: VGPR r -> row m0+r (lanes 0-15) / m0+r+8 (lanes 16-31)
    float* Dp = act + (size_t)(m0 + 8 * hs) * kBC + n0 + lr;
#pragma unroll
    for (int r = 0; r < 8; ++r) {
        float* dr = Dp + (size_t)r * kBC;
        dr[0]  = acc0[r];
        dr[16] = acc1[r];
        dr[32] = acc2[r];
        dr[48] = acc3[r];
    }
}

// ---------------------------------------------------------------------------
// Kernel 2: for each b (one block), stage act[:, b, :] (256x16 f32 = 16 KB) in
// LDS, then thread j computes (sum_i exp(-L1(act_i, act_j)) - 1) / 255 and
// writes out[j, 1024 + b].
// ---------------------------------------------------------------------------
__global__ __launch_bounds__(256) void pairwise_expl1(const float* __restrict__ act,
                                                      float* __restrict__ out) {
    __shared__ float4 tile[kN][4];  // 16 KB, row stride 64 B (b128-aligned)

    const int b = blockIdx.x;   // 0..127
    const int t = threadIdx.x;  // j = 0..255

    const float4* row = reinterpret_cast<const float4*>(act + (size_t)t * kBC + b * kC);
    const float4 m0 = row[0], m1 = row[1], m2 = row[2], m3 = row[3];
    tile[t][0] = m0; tile[t][1] = m1; tile[t][2] = m2; tile[t][3] = m3;
    __syncthreads();

    float acc = 0.f;
#pragma unroll 4
    for (int i = 0; i < kN; ++i) {
        const float4 v0 = tile[i][0];
        const float4 v1 = tile[i][1];
        const float4 v2 = tile[i][2];
        const float4 v3 = tile[i][3];
        float s;
        s  = fabsf(v0.x - m0.x) + fabsf(v0.y - m0.y) + fabsf(v0.z - m0.z) + fabsf(v0.w - m0.w);
        s += fabsf(v1.x - m1.x) + fabsf(v1.y - m1.y) + fabsf(v1.z - m1.z) + fabsf(v1.w - m1.w);
        s += fabsf(v2.x - m2.x) + fabsf(v2.y - m2.y) + fabsf(v2.z - m2.z) + fabsf(v2.w - m2.w);
        s += fabsf(v3.x - m3.x) + fabsf(v3.y - m3.y) + fabsf(v3.z - m3.z) + fabsf(v3.w - m3.w);
        acc += __expf(-s);
    }
    // self term i==j contributes exp(0)=1; subtract it, then mean over N-1
    out[(size_t)t * kOutW + kA + b] = (acc - 1.0f) * (1.0f / 255.0f);
}

// ---------------------------------------------------------------------------
// Kernel 3: out[:, 0:1024] = inp  (float4 copy; both strides 16B-aligned)
// grid 256 x 256 threads, one float4 per thread.
// ---------------------------------------------------------------------------
__global__ __launch_bounds__(256) void copy_inp(const float* __restrict__ inp,
                                                float* __restrict__ out) {
    const int idx = blockIdx.x * 256 + threadIdx.x;  // 0..65535
    const int r   = idx >> 8;                        // row 0..255
    const int c4  = idx & 255;                       // float4 col 0..255
    const float4 v = reinterpret_cast<const float4*>(inp + (size_t)r * kA)[c4];
    reinterpret_cast<float4*>(out + (size_t)r * kOutW)[c4] = v;
}

extern "C" void kernel_launch(void* const* d_in, const int* in_sizes, int n_in,
                              void* d_out, int out_size, void* d_ws, size_t ws_size,
                              hipStream_t stream) {
    const float* inp   = (const float*)d_in[0];   // [256, 1024] f32
    const float* theta = (const float*)d_in[1];   // [1024, 128, 16] f32
    float* out = (float*)d_out;                   // [256, 1152] f32
    float* act = (float*)d_ws;                    // [256, 2048] f32 scratch (2 MB)

    dim3 g1(kBC / 512, kN / 16);  // (4, 16)
    gemm_f32_wmma<<<g1, 256, 0, stream>>>(inp, theta, act);
    pairwise_expl1<<<kB, 256, 0, stream>>>(act, out);
    copy_inp<<<kN, 256, 0, stream>>>(inp, out);
}